// MyRNN_33921651704149
// MI455X (gfx1250) — compile-verified
//
#include <hip/hip_runtime.h>
#include <hip/hip_bf16.h>

// Problem constants (from reference): T=128, B=64, V=8192, H=2048
#define T_STEPS 128
#define B_SZ    64
#define V_SZ    8192
#define H_SZ    2048
#define VH      (V_SZ + H_SZ)   // 10240, hidden_w row stride

typedef __attribute__((ext_vector_type(2))) float v2f;
typedef __attribute__((ext_vector_type(8))) float v8f;

// ---------------------------------------------------------------------------
// Phase 1: one recurrent step.
//   new_state[b,h] = tanh( hidden_w[h, tok[b]]             (one-hot gather)
//                        + sum_k prev[b,k]*hidden_w[h,V+k] (WMMA f32 GEMM)
//                        + hidden_b[h] )
// One wave computes one 16x16 (b,h) tile. 512 waves total.
// prev == nullptr for t==0 (state0 = 0): skip the GEMM.
// ---------------------------------------------------------------------------
__global__ __launch_bounds__(128)
void rnn_step_kernel(const float* __restrict__ prev,       // (B,H) or nullptr
                     const float* __restrict__ hidden_w,   // (H, V+H)
                     const float* __restrict__ hidden_b,   // (H)
                     const int*   __restrict__ tok,        // (B) token ids of this step
                     float*       __restrict__ out_state)  // (B,H)
{
    const int lane = threadIdx.x & 31;
    const int wave = (blockIdx.x * blockDim.x + threadIdx.x) >> 5;
    const int tiles_h = H_SZ / 16;            // 128
    const int th = wave % tiles_h;
    const int tb = wave / tiles_h;            // 0..3
    const int h0 = th * 16;
    const int b0 = tb * 16;

    const int half = lane >> 4;               // 0: lanes 0-15, 1: lanes 16-31
    const int l16  = lane & 15;

    v8f acc = {};

    if (prev != nullptr) {
        // A fragment: prev[b0+l16, k + 2*half + {0,1}]        (16x4 f32, 2 VGPRs)
        const float* aPtr = prev + (size_t)(b0 + l16) * H_SZ + 2 * half;
        // B fragment: hidden_w[h0+l16, V + k + 2*half + {0,1}]  (B[k,n] = Whh[n,k])
        const float* bPtr = hidden_w + (size_t)(h0 + l16) * VH + V_SZ + 2 * half;

        #pragma unroll 8
        for (int k = 0; k < H_SZ; k += 4) {
            v2f a = *(const v2f*)(aPtr + k);
            v2f b = *(const v2f*)(bPtr + k);
            acc = __builtin_amdgcn_wmma_f32_16x16x4_f32(
                false, a, false, b, (short)0, acc, false, false);
        }
    }

    // Epilogue: lane holds C[m = r + 8*half, n = l16] for r = 0..7
    const int    h    = h0 + l16;
    const float  bias = hidden_b[h];
    const size_t hrow = (size_t)h * VH;

    #pragma unroll
    for (int r = 0; r < 8; ++r) {
        const int b   = b0 + r + 8 * half;
        const int t   = tok[b];                       // one-hot column index
        const float e = hidden_w[hrow + t];           // embedding gather
        out_state[(size_t)b * H_SZ + h] = tanhf(acc[r] + bias + e);
    }
}

// ---------------------------------------------------------------------------
// Phase 2: big output GEMM over ALL timesteps at once.
//   out[m, v] = sum_h S[m,h] * output_w[v,h] + output_b[v],  m = t*B + b
// M = T*B = 8192, N = V = 8192, K = H = 2048.
// Each wave computes a 64x64 output tile (4x4 WMMA tiles -> 16 independent
// accumulation chains per K-step; compiler auto-unrolls the K-loop 4x for a
// 64-WMMA / 32-load software pipeline). Operands stay L2-resident
// (output_w 64MB + states 64MB < 192MB L2); the 256MB result stream is
// written with non-temporal stores so it does not evict them.
// ---------------------------------------------------------------------------
__global__ __launch_bounds__(256)
void out_gemm_kernel(const float* __restrict__ S,         // (T*B, H) states
                     const float* __restrict__ output_w,  // (V, H)
                     const float* __restrict__ output_b,  // (V)
                     float*       __restrict__ out)       // (T*B, V)
{
    const int lane = threadIdx.x & 31;
    const int wave = (blockIdx.x * blockDim.x + threadIdx.x) >> 5;

    const int waves_n = V_SZ / 64;            // 128
    const int wn = wave % waves_n;
    const int wm = wave / waves_n;            // 0..127
    const int m0 = wm * 64;
    const int n0 = wn * 64;

    const int half = lane >> 4;
    const int l16  = lane & 15;

    const float* aBase = S        + (size_t)(m0 + l16) * H_SZ + 2 * half;
    const float* bBase = output_w + (size_t)(n0 + l16) * H_SZ + 2 * half;

    v8f acc[4][4];
    #pragma unroll
    for (int i = 0; i < 4; ++i)
        #pragma unroll
        for (int j = 0; j < 4; ++j)
            acc[i][j] = (v8f){};

    for (int k = 0; k < H_SZ; k += 4) {
        v2f a[4], b[4];
        #pragma unroll
        for (int i = 0; i < 4; ++i)
            a[i] = *(const v2f*)(aBase + (size_t)(16 * i) * H_SZ + k);
        #pragma unroll
        for (int j = 0; j < 4; ++j)
            b[j] = *(const v2f*)(bBase + (size_t)(16 * j) * H_SZ + k);

        #pragma unroll
        for (int i = 0; i < 4; ++i)
            #pragma unroll
            for (int j = 0; j < 4; ++j)
                acc[i][j] = __builtin_amdgcn_wmma_f32_16x16x4_f32(
                    false, a[i], false, b[j], (short)0, acc[i][j], false, false);
    }

    // Epilogue: C[m = r + 8*half, n = l16] per 16x16 tile; add output bias.
    // Non-temporal stores: the 256MB output is never re-read.
    #pragma unroll
    for (int j = 0; j < 4; ++j) {
        const int   n    = n0 + 16 * j + l16;
        const float bias = output_b[n];
        #pragma unroll
        for (int i = 0; i < 4; ++i) {
            const int mbase = m0 + 16 * i + 8 * half;
            #pragma unroll
            for (int r = 0; r < 8; ++r) {
                __builtin_nontemporal_store(acc[i][j][r] + bias,
                                            &out[(size_t)(mbase + r) * V_SZ + n]);
            }
        }
    }
}

// ---------------------------------------------------------------------------
// Copy final state (states of step T-1) to the tail of d_out.
// ---------------------------------------------------------------------------
__global__ __launch_bounds__(256)
void copy_final_kernel(const float* __restrict__ src, float* __restrict__ dst)
{
    const int i = blockIdx.x * blockDim.x + threadIdx.x;   // exactly B*H threads
    __builtin_nontemporal_store(src[i], &dst[i]);
}

extern "C" void kernel_launch(void* const* d_in, const int* in_sizes, int n_in,
                              void* d_out, int out_size, void* d_ws, size_t ws_size,
                              hipStream_t stream) {
    const int*   inputs   = (const int*)  d_in[0];   // (T, B) token ids
    const float* hidden_w = (const float*)d_in[1];   // (H, V+H)
    const float* hidden_b = (const float*)d_in[2];   // (H)
    const float* output_w = (const float*)d_in[3];   // (V, H)
    const float* output_b = (const float*)d_in[4];   // (V)
    float*       out      = (float*)d_out;           // T*B*V outputs ++ B*H final state

    float* states = (float*)d_ws;                    // (T, B, H) fp32 = 64 MB

    // Phase 1: sequential recurrence (graph-captured launches).
    for (int t = 0; t < T_STEPS; ++t) {
        const float* prev = (t == 0) ? nullptr : states + (size_t)(t - 1) * B_SZ * H_SZ;
        rnn_step_kernel<<<128, 128, 0, stream>>>(
            prev, hidden_w, hidden_b, inputs + (size_t)t * B_SZ,
            states + (size_t)t * B_SZ * H_SZ);
    }

    // Phase 2: one big GEMM for all T*B rows. 16384 waves of 64x64 tiles.
    out_gemm_kernel<<<2048, 256, 0, stream>>>(states, output_w, output_b, out);

    // Final state -> tail of d_out.
    copy_final_kernel<<<(B_SZ * H_SZ) / 256, 256, 0, stream>>>(
        states + (size_t)(T_STEPS - 1) * B_SZ * H_SZ,
        out + (size_t)T_STEPS * B_SZ * V_SZ);
}